// NonMaxSuppressionONNX_32787780338361
// MI455X (gfx1250) — compile-verified
//
#include <hip/hip_runtime.h>
#include <cstdint>
#include <cstddef>

// ---------------------------------------------------------------------------
// RPN proposal filtering + NMS for MI455X (gfx1250, wave32, 320KB LDS/WGP)
//
// Levels: n = {196608,49152,12288,3072,768}, pre-NMS topk per level
// k = {2000,2000,2000,2000,768}  ->  M = 8768 candidates total.
// Output: 1000 boxes (4 floats each) followed by 1000 scores = 5000 floats.
// ---------------------------------------------------------------------------

#define NLEV   5
#define MTOT   8768
#define NW     274          // ceil(8768/32); 274*32 == 8768 exactly
#define SORTN  16384        // padded bitonic size
#define POSTK  1000
#define IMGW   1024.0f
#define NEGINF (-__builtin_huge_valf())

// workspace layout (bytes)
#define OFF_HIST1   ((size_t)0)
#define OFF_HIST2   ((size_t)1310720)
#define OFF_THR     ((size_t)2621440)   // int thr[5*6]: T1,nA1,need2,T2,rem2,pad
#define OFF_CNT     ((size_t)2621696)   // uint cnt[10]: sel[5], tie[5]
#define OFF_CSCORE  ((size_t)2621952)
#define OFF_CBOX    ((size_t)2657024)
#define OFF_CLVL    ((size_t)2797312)
#define OFF_CVALID  ((size_t)2832384)
#define OFF_SSCORE  ((size_t)2867456)
#define OFF_SBOX    ((size_t)2902528)
#define OFF_SLVL    ((size_t)3042816)
#define OFF_SVALID  ((size_t)3077888)
#define OFF_SUP     ((size_t)3112960)
#define ZERO_BYTES  ((size_t)2621952)   // hist1 + hist2 + thr + cnt regions

__device__ __forceinline__ uint32_t fkey(float f) {
    uint32_t u = __float_as_uint(f);
    return (u & 0x80000000u) ? ~u : (u | 0x80000000u);
}

// -------------------------------------------------- 0) zero scratch regions
__global__ void zero_kernel(uint32_t* p, int nwords) {
    int i = blockIdx.x * blockDim.x + threadIdx.x;
    int stride = gridDim.x * blockDim.x;
    for (; i < nwords; i += stride) p[i] = 0u;
}

// -------------------------------------------------- 1) coarse histogram (top 16 bits)
__global__ void hist1_kernel(const float* __restrict__ logits, int n,
                             uint32_t* __restrict__ h) {
    int e = blockIdx.x * blockDim.x + threadIdx.x;
    if (e < n) atomicAdd(&h[fkey(logits[e]) >> 16], 1u);
}

// -------------------------------------------------- 2/4) threshold scan (shared by both passes)
__global__ void thresh_kernel(const uint32_t* __restrict__ hist,
                              int* __restrict__ thr, int pass) {
    int lvl = blockIdx.x;
    const uint32_t* h = hist + (size_t)lvl * 65536;
    int target = (pass == 0) ? ((lvl == 4) ? 768 : 2000) : thr[lvl * 6 + 2];

    __shared__ uint32_t csum[1024];
    uint32_t s = 0;
    int c0 = threadIdx.x * 64;
    for (int b = 0; b < 64; b++) s += h[c0 + b];
    csum[threadIdx.x] = s;
    __syncthreads();

    if (threadIdx.x == 0) {
        int cum = 0, T = 0, rem = target;
        for (int c = 1023; c >= 0; c--) {
            if (cum + (int)csum[c] >= target) {
                for (int b = c * 64 + 63; b >= c * 64; b--) {
                    int hb = (int)h[b];
                    if (cum + hb >= target) { T = b; rem = target - cum; break; }
                    cum += hb;
                }
                break;
            }
            cum += (int)csum[c];
        }
        if (pass == 0) {
            thr[lvl * 6 + 0] = T;
            thr[lvl * 6 + 1] = cum;           // strictly-above count
            thr[lvl * 6 + 2] = rem;           // still needed from bin T
        } else {
            thr[lvl * 6 + 3] = T;
            thr[lvl * 6 + 4] = rem;           // ties to take from (T1,T2) bin
        }
    }
}

// -------------------------------------------------- 3) refine histogram (low 16 bits within T1)
__global__ void hist2_kernel(const float* __restrict__ logits, int n,
                             uint32_t* __restrict__ h2,
                             const int* __restrict__ thr, int lvl) {
    int e = blockIdx.x * blockDim.x + threadIdx.x;
    if (e >= n) return;
    uint32_t key = fkey(logits[e]);
    if ((int)(key >> 16) == thr[lvl * 6 + 0])
        atomicAdd(&h2[key & 0xFFFFu], 1u);
}

// -------------------------------------------------- 5) compact exact top-k, clip, validity
__global__ void compact_kernel(const float* __restrict__ props,
                               const float* __restrict__ logits, int n,
                               int lvl, int base,
                               const int* __restrict__ thr,
                               unsigned* __restrict__ cnt,
                               float* __restrict__ cscore,
                               float4* __restrict__ cbox,
                               int* __restrict__ clvl,
                               int* __restrict__ cvalid) {
    int e = blockIdx.x * blockDim.x + threadIdx.x;
    if (e >= n) return;
    float sc = logits[e];
    uint32_t key = fkey(sc);
    uint32_t t16 = key >> 16, l16 = key & 0xFFFFu;
    uint32_t T1 = (uint32_t)thr[lvl * 6 + 0];
    uint32_t T2 = (uint32_t)thr[lvl * 6 + 3];
    int rem2 = thr[lvl * 6 + 4];

    bool sel = false;
    if (t16 > T1) sel = true;
    else if (t16 == T1) {
        if (l16 > T2) sel = true;
        else if (l16 == T2) {
            unsigned t = atomicAdd(&cnt[5 + lvl], 1u);
            sel = (t < (unsigned)rem2);
        }
    }
    if (!sel) return;

    unsigned s = atomicAdd(&cnt[lvl], 1u);
    int o = base + (int)s;
    float x1 = props[e * 4 + 0], y1 = props[e * 4 + 1];
    float x2 = props[e * 4 + 2], y2 = props[e * 4 + 3];
    x1 = fminf(fmaxf(x1, 0.f), IMGW);  y1 = fminf(fmaxf(y1, 0.f), IMGW);
    x2 = fminf(fmaxf(x2, 0.f), IMGW);  y2 = fminf(fmaxf(y2, 0.f), IMGW);
    cscore[o]  = sc;
    cbox[o]    = make_float4(x1, y1, x2, y2);
    clvl[o]    = lvl;
    cvalid[o]  = (((x2 - x1) > 0.f) && ((y2 - y1) > 0.f)) ? 1 : 0;
}

// -------------------------------------------------- 6) single-block bitonic sort (descending)
// 16384 x u64 keys = 128KB LDS -- fits CDNA5's 320KB WGP LDS in one workgroup.
__global__ __launch_bounds__(1024, 1)
void sort_kernel(const float* __restrict__ cscore, const float4* __restrict__ cbox,
                 const int* __restrict__ clvl, const int* __restrict__ cvalid,
                 float* __restrict__ sscore, float4* __restrict__ sbox,
                 int* __restrict__ slvl, int* __restrict__ svalid) {
    __shared__ uint64_t skey[SORTN];
    int tid = threadIdx.x;

    for (int i = tid; i < SORTN; i += 1024) {
        if (i < MTOT) {
            float ms = cvalid[i] ? cscore[i] : NEGINF;
            skey[i] = ((uint64_t)fkey(ms) << 32) | (uint32_t)i;
        } else {
            skey[i] = 0ull;   // sorts strictly below any real key (incl. -inf)
        }
    }
    __syncthreads();

    for (int k = 2; k <= SORTN; k <<= 1) {
        for (int j = k >> 1; j > 0; j >>= 1) {
            for (int i = tid; i < SORTN; i += 1024) {
                int ixj = i ^ j;
                if (ixj > i) {
                    uint64_t a = skey[i], b = skey[ixj];
                    bool desc = ((i & k) == 0);
                    if ((a < b) == desc) { skey[i] = b; skey[ixj] = a; }
                }
            }
            __syncthreads();
        }
    }

    for (int s = tid; s < MTOT; s += 1024) {
        uint32_t ci = (uint32_t)(skey[s] & 0xFFFFFFFFull);
        int v = cvalid[ci];
        sscore[s] = v ? cscore[ci] : NEGINF;
        sbox[s]   = cbox[ci];
        slvl[s]   = clvl[ci];
        svalid[s] = v;
    }
}

// -------------------------------------------------- 7) suppression bitmask, 1-wave blocks
// Upper-triangular tiles only (reduce never reads words left of the diagonal).
// Column tile of 32 boxes staged to LDS via CDNA5 async global->LDS DMA.
__global__ __launch_bounds__(32)
void mask_kernel(const float4* __restrict__ sbox, const int* __restrict__ slvl,
                 uint32_t* __restrict__ sup) {
    int cblk = blockIdx.x, rblk = blockIdx.y;
    if (cblk < rblk) return;              // lower triangle never consumed

    __shared__ float4 cb[32];
    __shared__ int    clv[32];
    int t = threadIdx.x;
    int j0 = cblk * 32;

    {   // async DMA: one b128 (box) + one b32 (level) per lane, no VGPR staging
        uint32_t ldsb = (uint32_t)(uintptr_t)(&cb[t]);
        unsigned long long gb = (unsigned long long)(uintptr_t)(sbox + j0 + t);
        asm volatile("global_load_async_to_lds_b128 %0, %1, off"
                     :: "v"(ldsb), "v"(gb) : "memory");
        uint32_t ldsl = (uint32_t)(uintptr_t)(&clv[t]);
        unsigned long long gl = (unsigned long long)(uintptr_t)(slvl + j0 + t);
        asm volatile("global_load_async_to_lds_b32 %0, %1, off"
                     :: "v"(ldsl), "v"(gl) : "memory");
        asm volatile("s_wait_asynccnt 0" ::: "memory");
        // single wave32 block: s_wait_asynccnt is sufficient, no barrier needed
    }

    int i = rblk * 32 + t;
    float4 bi = sbox[i];
    int li = slvl[i];
    float ai = (bi.z - bi.x) * (bi.w - bi.y);

    uint32_t bits = 0;
    for (int jj = 0; jj < 32; jj++) {
        int j = j0 + jj;
        if (j <= i) continue;                 // only j > i can be suppressed by i
        if (clv[jj] != li) continue;          // level offset => cross-level IoU == 0
        float4 bj = cb[jj];
        float xx1 = fmaxf(bi.x, bj.x), yy1 = fmaxf(bi.y, bj.y);
        float xx2 = fminf(bi.z, bj.z), yy2 = fminf(bi.w, bj.w);
        float iw = fmaxf(xx2 - xx1, 0.f), ih = fmaxf(yy2 - yy1, 0.f);
        float inter = iw * ih;
        float aj = (bj.z - bj.x) * (bj.w - bj.y);
        float iou = inter / (ai + aj - inter);
        if (iou > 0.7f) bits |= (1u << jj);
    }
    sup[(size_t)i * NW + cblk] = bits;
}

// -------------------------------------------------- 8) blocked sequential reduce + output
// 274 word-blocks: lane 0 resolves the 32x32 diagonal serially from LDS-staged
// diag words; kept rows are applied to the remaining words batched (one AND per
// word, OR-combining up to 32 sup rows). 2 barriers per block instead of 32.
__global__ __launch_bounds__(1024)
void nms_reduce_kernel(const uint32_t* __restrict__ sup,
                       const int* __restrict__ svalid,
                       const float4* __restrict__ sbox,
                       const float* __restrict__ sscore,
                       float* __restrict__ out) {
    __shared__ uint32_t keep[NW];
    __shared__ uint32_t diag[32];
    __shared__ int wpref[NW + 1];

    for (int w = threadIdx.x; w < NW; w += blockDim.x) {
        uint32_t bits = 0;
        for (int b = 0; b < 32; b++)
            bits |= (svalid[w * 32 + b] ? 1u : 0u) << b;
        keep[w] = bits;
    }
    __syncthreads();

    for (int wb = 0; wb < NW; wb++) {
        // stage this block's diagonal sup words into LDS (one wave32)
        if (threadIdx.x < 32)
            diag[threadIdx.x] = sup[(size_t)(wb * 32 + threadIdx.x) * NW + wb];
        __syncthreads();
        // exact sequential greedy within the 32-row block (lane 0, LDS-latency)
        if (threadIdx.x == 0) {
            uint32_t kw = keep[wb];
            for (int b = 0; b < 32; b++)
                if ((kw >> b) & 1u) kw &= ~diag[b];
            keep[wb] = kw;
        }
        __syncthreads();
        uint32_t kw = keep[wb];                 // final keep bits of this block
        if (kw) {
            for (int w = wb + 1 + (int)threadIdx.x; w < NW; w += (int)blockDim.x) {
                uint32_t m = 0, kk = kw;
                while (kk) {
                    int b = __ffs(kk) - 1;
                    kk &= kk - 1u;
                    m |= sup[(size_t)(wb * 32 + b) * NW + w];
                }
                keep[w] &= ~m;
            }
        }
    }
    __syncthreads();

    if (threadIdx.x == 0) {
        int cum = 0;
        for (int w = 0; w < NW; w++) { wpref[w] = cum; cum += __popc(keep[w]); }
        wpref[NW] = cum;
    }
    __syncthreads();
    int nK = wpref[NW];
    for (int p = threadIdx.x; p < MTOT; p += blockDim.x) {
        int w = p >> 5, b = p & 31;
        uint32_t kw = keep[w];
        int kr = wpref[w] + __popc(kw & ((1u << b) - 1u));
        bool kp = (kw >> b) & 1u;
        // top_k over (kept scores desc, then -inf ties by lowest position)
        int r = kp ? kr : (nK + (p - kr));
        if (r < POSTK) {
            float4 bx = sbox[p];
            out[r * 4 + 0] = bx.x; out[r * 4 + 1] = bx.y;
            out[r * 4 + 2] = bx.z; out[r * 4 + 3] = bx.w;
            out[POSTK * 4 + r] = kp ? sscore[p] : NEGINF;
        }
    }
}

// ---------------------------------------------------------------------------
extern "C" void kernel_launch(void* const* d_in, const int* in_sizes, int n_in,
                              void* d_out, int out_size, void* d_ws, size_t ws_size,
                              hipStream_t stream) {
    (void)n_in; (void)out_size; (void)ws_size;
    char* ws = (char*)d_ws;
    uint32_t* hist1 = (uint32_t*)(ws + OFF_HIST1);
    uint32_t* hist2 = (uint32_t*)(ws + OFF_HIST2);
    int*      thr   = (int*)     (ws + OFF_THR);
    unsigned* cnt   = (unsigned*)(ws + OFF_CNT);
    float*    cscore= (float*)   (ws + OFF_CSCORE);
    float4*   cbox  = (float4*)  (ws + OFF_CBOX);
    int*      clvl  = (int*)     (ws + OFF_CLVL);
    int*      cvalid= (int*)     (ws + OFF_CVALID);
    float*    sscore= (float*)   (ws + OFF_SSCORE);
    float4*   sbox  = (float4*)  (ws + OFF_SBOX);
    int*      slvl  = (int*)     (ws + OFF_SLVL);
    int*      svalid= (int*)     (ws + OFF_SVALID);
    uint32_t* sup   = (uint32_t*)(ws + OFF_SUP);
    float*    out   = (float*)d_out;

    const int kbase[NLEV] = {0, 2000, 4000, 6000, 8000};

    // 0) zero histograms + counters
    {
        int nwords = (int)(ZERO_BYTES / 4);
        zero_kernel<<<(nwords + 255) / 256, 256, 0, stream>>>((uint32_t*)ws, nwords);
    }
    // 1) coarse histograms
    for (int lvl = 0; lvl < NLEV; lvl++) {
        const float* logits = (const float*)d_in[2 * lvl + 1];
        int n = in_sizes[2 * lvl + 1];
        hist1_kernel<<<(n + 255) / 256, 256, 0, stream>>>(
            logits, n, hist1 + (size_t)lvl * 65536);
    }
    // 2) coarse thresholds
    thresh_kernel<<<NLEV, 1024, 0, stream>>>(hist1, thr, 0);
    // 3) refine histograms
    for (int lvl = 0; lvl < NLEV; lvl++) {
        const float* logits = (const float*)d_in[2 * lvl + 1];
        int n = in_sizes[2 * lvl + 1];
        hist2_kernel<<<(n + 255) / 256, 256, 0, stream>>>(
            logits, n, hist2 + (size_t)lvl * 65536, thr, lvl);
    }
    // 4) refined thresholds (exact 32-bit key cut)
    thresh_kernel<<<NLEV, 1024, 0, stream>>>(hist2, thr, 1);
    // 5) compact exact per-level top-k
    for (int lvl = 0; lvl < NLEV; lvl++) {
        const float* props  = (const float*)d_in[2 * lvl];
        const float* logits = (const float*)d_in[2 * lvl + 1];
        int n = in_sizes[2 * lvl + 1];
        compact_kernel<<<(n + 255) / 256, 256, 0, stream>>>(
            props, logits, n, lvl, kbase[lvl], thr, cnt,
            cscore, cbox, clvl, cvalid);
    }
    // 6) global descending sort of 8768 candidates (one workgroup, 128KB LDS)
    sort_kernel<<<1, 1024, 0, stream>>>(cscore, cbox, clvl, cvalid,
                                        sscore, sbox, slvl, svalid);
    // 7) suppression bitmask (single-wave blocks, async LDS staging, upper tri)
    mask_kernel<<<dim3(NW, NW), 32, 0, stream>>>(sbox, slvl, sup);
    // 8) blocked keep-reduction + final top-1000 emission
    nms_reduce_kernel<<<1, 1024, 0, stream>>>(sup, svalid, sbox, sscore, out);
}